// DenseExpert_25898652794998
// MI455X (gfx1250) — compile-verified
//
#include <hip/hip_runtime.h>

// Problem constants (match the JAX reference)
#define B_     32768
#define IN_    1024
#define UNITS_ 1024
#define E_     8

typedef float v2f __attribute__((ext_vector_type(2)));
typedef float v4f __attribute__((ext_vector_type(4)));
typedef float v8f __attribute__((ext_vector_type(8)));
typedef unsigned int u32x4 __attribute__((ext_vector_type(4)));
typedef int i32x4 __attribute__((ext_vector_type(4)));
typedef int i32x8 __attribute__((ext_vector_type(8)));

// Tiling
#define BM  128
#define BN  128
#define BK  16
#define LDA 20     // A tile LDS stride (floats): 16 data + 4 pad (TDM pad 3/3)
#define LDB 136    // B tile LDS stride (floats): 128 data + 8 pad (TDM pad 6/7)
#define SB_EXPERT (BK * LDB)         // 2176 floats per expert plane
#define SB_BUF    (E_ * SB_EXPERT)   // 17408 floats per buffer
#define SA_BUF    (BM * LDA)         // 2560 floats per buffer

// ---- Tensor Data Mover: build D# per CDNA5 ISA and issue ----
// All dims/strides in data_size (=4B) element units.
// Device-only: the builtin is not registered in the host pass.
__device__ __forceinline__ void tdm_load(unsigned lds_off,
                                         unsigned long long gaddr,
                                         unsigned td0, unsigned td1, unsigned td2,
                                         unsigned tl0, unsigned tl1, unsigned tl2,
                                         unsigned long long s0, unsigned long long s1,
                                         unsigned pad_interval, unsigned pad_amount) {
#if defined(__HIP_DEVICE_COMPILE__)
  u32x4 g0;
  g0[0] = 1u;                                   // count=1 (valid user descriptor)
  g0[1] = lds_off;                              // LDS byte address
  g0[2] = (unsigned)(gaddr & 0xFFFFFFFFull);    // global_addr[31:0]
  g0[3] = (unsigned)((gaddr >> 32) & 0x01FFFFFFull) | (2u << 30); // addr[56:32], type=2

  i32x8 g1;
  g1[0] = (int)((2u << 16)                      // data_size = 4 bytes
              | (1u << 20)                      // pad_enable
              | (pad_interval << 22)
              | (pad_amount << 25));
  g1[1] = (int)((td0 & 0xFFFFu) << 16);                         // tensor_dim0 lo
  g1[2] = (int)((td0 >> 16) | ((td1 & 0xFFFFu) << 16));         // dim0 hi | dim1 lo
  g1[3] = (int)((td1 >> 16) | (tl0 << 16));                     // dim1 hi | tile_dim0
  g1[4] = (int)(tl1 | (tl2 << 16));                             // tile_dim1 | tile_dim2
  g1[5] = (int)(s0 & 0xFFFFFFFFull);                            // dim0_stride lo32
  g1[6] = (int)(((s0 >> 32) & 0xFFFFull) | ((s1 & 0xFFFFull) << 16)); // s0 hi | s1 lo16
  g1[7] = (int)((s1 >> 16) & 0xFFFFFFFFull);                    // dim1_stride [47:16]

  i32x4 g2;
  g2[0] = (int)td2;   // tensor_dim2
  g2[1] = 0;          // tensor_dim3
  g2[2] = 0;          // tensor_dim2_stride lo (unused: tile_dim3=0)
  g2[3] = 0;          // stride hi | tile_dim3=0
  i32x4 g3;
  g3[0] = 0; g3[1] = 0; g3[2] = 0; g3[3] = 0;

  i32x8 g4;           // trailing group (unused for <=3D tensors): zero-filled
  g4[0] = 0; g4[1] = 0; g4[2] = 0; g4[3] = 0;
  g4[4] = 0; g4[5] = 0; g4[6] = 0; g4[7] = 0;

  __builtin_amdgcn_tensor_load_to_lds(g0, g1, g2, g3, g4, 0);
#else
  (void)lds_off; (void)gaddr; (void)td0; (void)td1; (void)td2;
  (void)tl0; (void)tl1; (void)tl2; (void)s0; (void)s1;
  (void)pad_interval; (void)pad_amount;
#endif
}

__device__ __forceinline__ void tdm_wait0() {
#if defined(__HIP_DEVICE_COMPILE__)
  __builtin_amdgcn_s_wait_tensorcnt(0);
#endif
}

__global__ __launch_bounds__(256)
void moe_dense_wmma_tdm(const float* __restrict__ x,
                        const float* __restrict__ gate,
                        const float* __restrict__ alpha,
                        const float* __restrict__ beta,
                        float* __restrict__ out) {
  // Double-buffered TDM-fed tiles: 20 KB (A) + 136 KB (B) = 156 KB of 320 KB LDS
  // -> two workgroups co-resident per WGP (312 KB), 4 waves/SIMD.
  __shared__ float sA[2 * SA_BUF];
  __shared__ float sB[2 * SB_BUF];

  const int tid  = threadIdx.x;
  const int lane = tid & 31;
  const int wave = tid >> 5;       // 0..7
  const int wm   = wave >> 1;      // 0..3 : 32-row band
  const int wn   = wave & 1;       // 0..1 : 64-col band

  const int tileM = (int)blockIdx.x * BM;
  const int tileN = (int)blockIdx.y * BN;

  // V_WMMA_F32_16X16X4_F32 fragment mapping (wave32):
  //  A: lane L -> M = L&15, Kb = (L>>4)*2, a = {A[M][Kb], A[M][Kb+1]}
  //  B: lane L -> N = L&15, Kb = (L>>4)*2, b = {B[Kb][N], B[Kb+1][N]}
  //  C: VGPR v -> M = v + 8*(L>>4), N = L&15
  const int fm = lane & 15;
  const int fk = (lane >> 4) * 2;

  // Per-lane gate weights for this lane's two A-subtile rows (expert mix),
  // statically indexed by expert in the unrolled main loop.
  float g[2][E_];
  #pragma unroll
  for (int i = 0; i < 2; ++i) {
    const size_t mg = (size_t)(tileM + wm * 32 + i * 16 + fm);
    v4f glo = *(const v4f*)(gate + mg * E_);
    v4f ghi = *(const v4f*)(gate + mg * E_ + 4);
    g[i][0] = glo.x; g[i][1] = glo.y; g[i][2] = glo.z; g[i][3] = glo.w;
    g[i][4] = ghi.x; g[i][5] = ghi.y; g[i][6] = ghi.z; g[i][7] = ghi.w;
  }

  v8f acc[2][4];
  #pragma unroll
  for (int i = 0; i < 2; ++i)
    #pragma unroll
    for (int j = 0; j < 4; ++j)
      #pragma unroll
      for (int v = 0; v < 8; ++v)
        acc[i][j][v] = 0.0f;

  const unsigned ldsA0 = (unsigned)(unsigned long long)(const void*)&sA[0];
  const unsigned ldsB0 = (unsigned)(unsigned long long)(const void*)&sB[0];
  const unsigned long long xBase = (unsigned long long)(const void*)
                                   (x + (size_t)tileM * IN_);
  const unsigned long long aBase = (unsigned long long)(const void*)
                                   (alpha + (size_t)tileN);

  const int nsteps = IN_ / BK;  // 64

  // Prologue: DMA first tiles into buffer 0 (one issuing wave; EXEC ignored by TDM)
  if (wave == 0) {
    // A: 2D tile 16(k) x 128(m) of x, pad 16 dwords -> +4 (stride 20)
    tdm_load(ldsA0, xBase,
             IN_, B_, 1, BK, BM, 0,
             IN_, 0, /*pad_interval(16dw)=*/3u, /*pad_amount(4dw)=*/3u);
    // B: 3D tile 128(u) x 16(k, stride IN) x 8(e, stride IN*UNITS), pad 128 -> +8 (stride 136)
    tdm_load(ldsB0, aBase,
             UNITS_, IN_, E_, BN, BK, E_,
             UNITS_, (unsigned long long)IN_ * UNITS_,
             /*pad_interval(128dw)=*/6u, /*pad_amount(8dw)=*/7u);
    tdm_wait0();
  }
  __syncthreads();

  for (int s = 0; s < nsteps; ++s) {
    const int cur = s & 1;
    const int nxt = cur ^ 1;

    // Kick off DMA for the next K-step into the other buffer (overlaps compute)
    if (wave == 0 && (s + 1) < nsteps) {
      const unsigned long long koff = (unsigned long long)(s + 1) * BK * 4ull;
      tdm_load(ldsA0 + (unsigned)(nxt * SA_BUF * 4), xBase + koff,
               IN_, B_, 1, BK, BM, 0,
               IN_, 0, 3u, 3u);
      tdm_load(ldsB0 + (unsigned)(nxt * SB_BUF * 4), aBase + koff * UNITS_,
               UNITS_, IN_, E_, BN, BK, E_,
               UNITS_, (unsigned long long)IN_ * UNITS_, 6u, 7u);
    }

    const float* A  = sA + cur * SA_BUF;
    const float* Bt = sB + cur * SB_BUF;

    // Raw A fragments for this K-step (shared across all experts)
    v2f araw[2][4];
    #pragma unroll
    for (int i = 0; i < 2; ++i)
      #pragma unroll
      for (int q = 0; q < 4; ++q)
        araw[i][q] = *(const v2f*)(A + (wm * 32 + i * 16 + fm) * LDA + q * 4 + fk);

    #pragma unroll
    for (int e = 0; e < E_; ++e) {
      const float* Be = Bt + e * SB_EXPERT;
      #pragma unroll
      for (int q = 0; q < 4; ++q) {
        v2f af[2];
        af[0] = araw[0][q] * g[0][e];   // fold gate[b,e] into the A operand
        af[1] = araw[1][q] * g[1][e];
        v2f bf[4];
        #pragma unroll
        for (int j = 0; j < 4; ++j) {
          const int n = wn * 64 + j * 16 + fm;
          bf[j].x = Be[(q * 4 + fk)     * LDB + n];
          bf[j].y = Be[(q * 4 + fk + 1) * LDB + n];
        }
        #pragma unroll
        for (int i = 0; i < 2; ++i)
          #pragma unroll
          for (int j = 0; j < 4; ++j)
            acc[i][j] = __builtin_amdgcn_wmma_f32_16x16x4_f32(
                false, af[i], false, bf[j],
                (short)0, acc[i][j], false, false);
      }
    }

    // Make sure next tiles landed, then release everyone into the next step
    if (wave == 0 && (s + 1) < nsteps)
      tdm_wait0();
    __syncthreads();
  }

  // ---- Epilogue: fold bias = gate @ beta (16x8 @ 8x16, K=8) in on the matrix
  // pipe: two more K=4 WMMAs per subtile, accumulating into the existing C regs.
  // Gate A-fragments are loaded directly in fragment layout (per-lane v2f) to
  // avoid dynamic indexing of the g[] register array by fk.
  v2f ag[2][2];
  #pragma unroll
  for (int i = 0; i < 2; ++i)
    #pragma unroll
    for (int q = 0; q < 2; ++q) {
      const size_t mg = (size_t)(tileM + wm * 32 + i * 16 + fm);
      ag[i][q] = *(const v2f*)(gate + mg * E_ + q * 4 + fk);
    }
  #pragma unroll
  for (int j = 0; j < 4; ++j) {
    const int ncol = tileN + wn * 64 + j * 16 + fm;
    #pragma unroll
    for (int q = 0; q < 2; ++q) {
      v2f bf;
      bf.x = beta[(size_t)(q * 4 + fk)     * UNITS_ + ncol];
      bf.y = beta[(size_t)(q * 4 + fk + 1) * UNITS_ + ncol];
      #pragma unroll
      for (int i = 0; i < 2; ++i)
        acc[i][j] = __builtin_amdgcn_wmma_f32_16x16x4_f32(
            false, ag[i][q], false, bf,
            (short)0, acc[i][j], false, false);
    }
  }

  // ---- Store (coalesced: lanes 0-15 / 16-31 cover 16 consecutive columns) ----
  const int cn  = lane & 15;
  const int cm0 = (lane >> 4) * 8;
  #pragma unroll
  for (int j = 0; j < 4; ++j) {
    const int ncol = tileN + wn * 64 + j * 16 + cn;
    #pragma unroll
    for (int i = 0; i < 2; ++i) {
      #pragma unroll
      for (int v = 0; v < 8; ++v) {
        const int mrow = tileM + wm * 32 + i * 16 + cm0 + v;
        out[(size_t)mrow * UNITS_ + ncol] = acc[i][j][v];
      }
    }
  }
}

extern "C" void kernel_launch(void* const* d_in, const int* in_sizes, int n_in,
                              void* d_out, int out_size, void* d_ws, size_t ws_size,
                              hipStream_t stream) {
  (void)in_sizes; (void)n_in; (void)d_ws; (void)ws_size; (void)out_size;
  const float* x     = (const float*)d_in[0];
  const float* gate  = (const float*)d_in[1];
  const float* alpha = (const float*)d_in[2];
  const float* beta  = (const float*)d_in[3];
  float* out = (float*)d_out;

  dim3 grid(B_ / BM, UNITS_ / BN, 1);  // 256 x 8
  dim3 block(256, 1, 1);               // 8 wave32
  moe_dense_wmma_tdm<<<grid, block, 0, stream>>>(x, gate, alpha, beta, out);
}